// pGNNNet_14053132993015
// MI455X (gfx1250) — compile-verified
//
#include <hip/hip_runtime.h>
#include <hip/hip_bf16.h>
#include <math.h>

// ---------------------------------------------------------------------------
// pGNNNet on MI455X (gfx1250, wave32).
//  P==2.0  =>  M_e = gnorm^(P-2) = 1 exactly, so the per-edge gradient-norm
//  pass in the reference is the identity; propagation reduces to a
//  symmetric-normalized SpMM:  agg[d] += f[s]/sqrt(deg_s*deg_d), then
//  f = alpha*agg + beta*h with alpha=1/(cnt/deg + 0.1), beta=0.1*alpha.
//  GEMMs use V_WMMA_F32_16X16X4_F32 (fp32 in/out, 16x16 tiles).
// ---------------------------------------------------------------------------

#define NN   100000
#define EE   1600000
#define IN_  256
#define HID_ 64
#define OUT_ 40
#define COEF 0.1f           // 2*MU/P = 2*0.1/2.0

typedef __attribute__((ext_vector_type(2))) float v2f;
typedef __attribute__((ext_vector_type(8))) float v8f;

// ---------------------------------------------------------------- utilities
__global__ void zero_f32_kernel(float* __restrict__ p, int n) {
    int i = blockIdx.x * blockDim.x + threadIdx.x;
    if (i < n) p[i] = 0.0f;
}

__global__ void zero_u32_kernel(unsigned* __restrict__ p, int n) {
    int i = blockIdx.x * blockDim.x + threadIdx.x;
    if (i < n) p[i] = 0u;
}

__global__ void count_deg_kernel(const int* __restrict__ dst,
                                 unsigned* __restrict__ cnt, int numEdges) {
    int e = blockIdx.x * blockDim.x + threadIdx.x;
    if (e < numEdges) atomicAdd(&cnt[dst[e]], 1u);
}

__global__ void node_coeff_kernel(const unsigned* __restrict__ cnt,
                                  float* __restrict__ dinvs,
                                  float* __restrict__ alpha, int n) {
    int i = blockIdx.x * blockDim.x + threadIdx.x;
    if (i >= n) return;
    float c   = (float)cnt[i];
    float deg = fmaxf(c, 1.0f);                 // deg = max(segment_sum, 1)
    dinvs[i]  = rsqrtf(deg);
    float denom = c / deg + COEF;               // segsum(M=1)/deg + 2mu/p
    alpha[i]  = 1.0f / denom;
}

// ------------------------------------------------- GEMM1: h = relu(x@W1+b1)
// One wave computes a 16-row x 64-col tile. K-loop: 256/4 = 64 WMMA steps,
// A fragment shared by the 4 N-tiles.
__global__ void gemm1_wmma_kernel(const float* __restrict__ X,
                                  const float* __restrict__ W1,
                                  const float* __restrict__ b1,
                                  float* __restrict__ H) {
    const int wavesPerBlock = blockDim.x >> 5;
    const int wave = blockIdx.x * wavesPerBlock + (threadIdx.x >> 5);
    const int tiles = NN / 16;                  // 6250 (exact)
    if (wave >= tiles) return;                  // wave-uniform exit
    const int lane    = threadIdx.x & 31;
    const int l16     = lane & 15;
    const int halfSel = lane >> 4;              // 0: lanes 0-15, 1: 16-31
    const int row0    = wave * 16;
    const int arow    = row0 + l16;

    v8f acc[4];
    #pragma unroll
    for (int t = 0; t < 4; ++t) acc[t] = (v8f){};

    for (int k0 = 0; k0 < IN_; k0 += 4) {
        // A fragment (ISA 7.12.2): lane<16 -> K=k0,k0+1 ; lane>=16 -> K=k0+2,k0+3
        const int ka = k0 + (halfSel << 1);
        v2f a = *(const v2f*)(X + (size_t)arow * IN_ + ka);
        #pragma unroll
        for (int t = 0; t < 4; ++t) {
            const int col = t * 16 + l16;
            v2f b;
            b.x = W1[(size_t)ka * HID_ + col];
            b.y = W1[(size_t)(ka + 1) * HID_ + col];
            acc[t] = __builtin_amdgcn_wmma_f32_16x16x4_f32(
                false, a, false, b, (short)0, acc[t], false, false);
        }
    }
    // bias + relu + store.  C layout: VGPR r -> rows (r, r+8) split by lane half.
    #pragma unroll
    for (int t = 0; t < 4; ++t) {
        const int col = t * 16 + l16;
        const float bv = b1[col];
        #pragma unroll
        for (int r = 0; r < 8; ++r) {
            const int row = row0 + r + (halfSel << 3);
            float v = acc[t][r] + bv;
            H[(size_t)row * HID_ + col] = v > 0.0f ? v : 0.0f;
        }
    }
}

// ------------------------------------------------------------------- SpMM
// agg[dst] += dinvs[src]*dinvs[dst] * f[src]; one wave per 4 edges (32 lanes x
// float2 = 64 channels), L2-resident working set (192MB L2 >> 90MB).
#define EDGES_PER_WAVE 4
__global__ void spmm_kernel(const int* __restrict__ src,
                            const int* __restrict__ dst,
                            const float* __restrict__ dinvs,
                            const float* __restrict__ fin,
                            float* __restrict__ agg, int numEdges) {
    const int lane = threadIdx.x & 31;
    const int wave = (blockIdx.x * blockDim.x + threadIdx.x) >> 5;
    int e0 = wave * EDGES_PER_WAVE;
    if (e0 >= numEdges) return;
    int eEnd = e0 + EDGES_PER_WAVE;
    if (eEnd > numEdges) eEnd = numEdges;
    const int ch = lane << 1;                    // 2 channels per lane
    for (int e = e0; e < eEnd; ++e) {
        const int s = src[e];
        const int d = dst[e];
        if (e + 1 < eEnd) {                      // global_prefetch_b8 next gather row
            const int sn = src[e + 1];
            __builtin_prefetch(fin + (size_t)sn * HID_ + ch, 0, 1);
        }
        const float c = dinvs[s] * dinvs[d];
        const float2 v = *(const float2*)(fin + (size_t)s * HID_ + ch);
        float* ap = agg + (size_t)d * HID_ + ch;
        atomicAdd(ap,     c * v.x);
        atomicAdd(ap + 1, c * v.y);
    }
}

// f = alpha*agg + beta*h  (in place on agg), beta = COEF*alpha
__global__ void finalize_kernel(float* __restrict__ f,
                                const float* __restrict__ h,
                                const float* __restrict__ alpha, int n) {
    int i = blockIdx.x * blockDim.x + threadIdx.x;
    if (i >= n) return;
    const float a = alpha[i >> 6];               // node = i / HID_
    f[i] = a * f[i] + (COEF * a) * h[i];
}

// --------------------------------------------- GEMM2: logits = f@W2 + b2
// OUT=40 padded to 48 (3 N-tiles of 16); cols >= 40 masked on store.
__global__ void gemm2_wmma_kernel(const float* __restrict__ F,
                                  const float* __restrict__ W2,
                                  const float* __restrict__ b2,
                                  float* __restrict__ Out) {
    const int wavesPerBlock = blockDim.x >> 5;
    const int wave = blockIdx.x * wavesPerBlock + (threadIdx.x >> 5);
    const int tiles = NN / 16;
    if (wave >= tiles) return;
    const int lane    = threadIdx.x & 31;
    const int l16     = lane & 15;
    const int halfSel = lane >> 4;
    const int row0    = wave * 16;
    const int arow    = row0 + l16;

    v8f acc[3];
    #pragma unroll
    for (int t = 0; t < 3; ++t) acc[t] = (v8f){};

    for (int k0 = 0; k0 < HID_; k0 += 4) {
        const int ka = k0 + (halfSel << 1);
        v2f a = *(const v2f*)(F + (size_t)arow * HID_ + ka);
        #pragma unroll
        for (int t = 0; t < 3; ++t) {
            const int col = t * 16 + l16;
            v2f b;
            b.x = (col < OUT_) ? W2[(size_t)ka * OUT_ + col] : 0.0f;
            b.y = (col < OUT_) ? W2[(size_t)(ka + 1) * OUT_ + col] : 0.0f;
            acc[t] = __builtin_amdgcn_wmma_f32_16x16x4_f32(
                false, a, false, b, (short)0, acc[t], false, false);
        }
    }
    #pragma unroll
    for (int t = 0; t < 3; ++t) {
        const int col = t * 16 + l16;
        if (col < OUT_) {                        // divergence only after all WMMAs
            const float bv = b2[col];
            #pragma unroll
            for (int r = 0; r < 8; ++r) {
                const int row = row0 + r + (halfSel << 3);
                Out[(size_t)row * OUT_ + col] = acc[t][r] + bv;
            }
        }
    }
}

// --------------------------------------------------- log_softmax (in place)
__global__ void log_softmax_kernel(float* __restrict__ Out) {
    const int wavesPerBlock = blockDim.x >> 5;
    const int row = blockIdx.x * wavesPerBlock + (threadIdx.x >> 5);
    if (row >= NN) return;
    const int lane = threadIdx.x & 31;
    float* rp = Out + (size_t)row * OUT_;
    const float NEG = -3.402823466e38f;
    float v0 = (lane < OUT_) ? rp[lane] : NEG;           // OUT_=40: lane & lane+32
    float v1 = (lane + 32 < OUT_) ? rp[lane + 32] : NEG;
    float m = fmaxf(v0, v1);
    #pragma unroll
    for (int off = 16; off > 0; off >>= 1) m = fmaxf(m, __shfl_xor(m, off, 32));
    float s = ((lane < OUT_) ? expf(v0 - m) : 0.0f) +
              ((lane + 32 < OUT_) ? expf(v1 - m) : 0.0f);
    #pragma unroll
    for (int off = 16; off > 0; off >>= 1) s += __shfl_xor(s, off, 32);
    const float lse = m + logf(s);
    if (lane < OUT_)      rp[lane]      = v0 - lse;
    if (lane + 32 < OUT_) rp[lane + 32] = v1 - lse;
}

// ---------------------------------------------------------------- launcher
extern "C" void kernel_launch(void* const* d_in, const int* in_sizes, int n_in,
                              void* d_out, int out_size, void* d_ws, size_t ws_size,
                              hipStream_t stream) {
    const float* x  = (const float*)d_in[0];          // [N, 256]
    const int*   ei = (const int*)  d_in[1];          // [2, E] flat
    const float* W1 = (const float*)d_in[2];          // [256, 64]
    const float* b1 = (const float*)d_in[3];          // [64]
    const float* W2 = (const float*)d_in[4];          // [64, 40]
    const float* b2 = (const float*)d_in[5];          // [40]
    float* out = (float*)d_out;                       // [N, 40]

    const int* src = ei;                              // edge_index[0]
    const int* dst = ei + EE;                         // edge_index[1]

    // workspace carve-up (~78 MB): h | fA | fB | cnt | dinvs | alpha
    const size_t nh = (size_t)NN * HID_;
    float*    hBuf  = (float*)d_ws;
    float*    fA    = hBuf + nh;
    float*    fB    = fA + nh;
    unsigned* cnt   = (unsigned*)(fB + nh);
    float*    dinvs = (float*)(cnt + NN);
    float*    alpha = dinvs + NN;

    const int B = 256;
    const int wavesPerBlock = B / 32;
    const int tiles = NN / 16;
    const int tileBlocks = (tiles + wavesPerBlock - 1) / wavesPerBlock;

    // degree + per-node coefficients
    zero_u32_kernel<<<(NN + B - 1) / B, B, 0, stream>>>(cnt, NN);
    count_deg_kernel<<<(EE + B - 1) / B, B, 0, stream>>>(dst, cnt, EE);
    node_coeff_kernel<<<(NN + B - 1) / B, B, 0, stream>>>(cnt, dinvs, alpha, NN);

    // h = relu(x @ W1 + b1)
    gemm1_wmma_kernel<<<tileBlocks, B, 0, stream>>>(x, W1, b1, hBuf);

    const int nhInt = (int)nh;
    const int waves = (EE + EDGES_PER_WAVE - 1) / EDGES_PER_WAVE;
    const int spmmBlocks = (waves + wavesPerBlock - 1) / wavesPerBlock;

    // iteration 1: fA = alpha * SpMM(h) + beta * h
    zero_f32_kernel<<<(nhInt + B - 1) / B, B, 0, stream>>>(fA, nhInt);
    spmm_kernel<<<spmmBlocks, B, 0, stream>>>(src, dst, dinvs, hBuf, fA, EE);
    finalize_kernel<<<(nhInt + B - 1) / B, B, 0, stream>>>(fA, hBuf, alpha, nhInt);

    // iteration 2: fB = alpha * SpMM(fA) + beta * h
    zero_f32_kernel<<<(nhInt + B - 1) / B, B, 0, stream>>>(fB, nhInt);
    spmm_kernel<<<spmmBlocks, B, 0, stream>>>(src, dst, dinvs, fA, fB, EE);
    finalize_kernel<<<(nhInt + B - 1) / B, B, 0, stream>>>(fB, hBuf, alpha, nhInt);

    // logits + log_softmax
    gemm2_wmma_kernel<<<tileBlocks, B, 0, stream>>>(fB, W2, b2, out);
    log_softmax_kernel<<<(NN + wavesPerBlock - 1) / wavesPerBlock, B, 0, stream>>>(out);
}